// _EfficientNonSamplingFactorizationMachines_67456756351037
// MI455X (gfx1250) — compile-verified
//
#include <hip/hip_runtime.h>

#define DIM 64
#define OUTW (DIM + 2)

typedef __attribute__((ext_vector_type(2))) float v2f;
typedef __attribute__((ext_vector_type(8))) float v8f;

// One wave (32 lanes) per output row; 16 waves (16 rows) per 512-thread block.
// MODE 0: user rows -> tail = [scalar + bias, 1]
// MODE 1: item rows -> tail = [1, scalar]
template <int L, int MODE>
__global__ __launch_bounds__(512)
void fm_pool_kernel(const float* __restrict__ emb,   // [V, 64]
                    const float* __restrict__ w,     // [V]
                    const int*   __restrict__ idx,   // [N, L]
                    const float* __restrict__ h1,    // [64]
                    const float* __restrict__ bias,  // [1] (used when MODE==0)
                    float* __restrict__ out)         // [N, 66]
{
    __shared__ float lds_bi[16][68];   // padded stride 68 -> conflict-free column reads
    __shared__ float lds_ws[16];
    __shared__ float lds_h1[64];

    const int tid  = threadIdx.x;
    const int lane = tid & 31;
    const int wv   = tid >> 5;                  // wave id = local row 0..15
    const int row  = blockIdx.x * 16 + wv;

    if (tid < 64) lds_h1[tid] = h1[tid];

    const float bias0 = (MODE == 0) ? bias[0] : 0.0f;

    // ---- index prefetch: 2 coalesced loads per row, broadcast via readlane ----
    const int* irow = idx + (size_t)row * L;
    const int i0 = (lane      < L) ? irow[lane]      : 0;
    const int i1 = (lane + 32 < L) ? irow[lane + 32] : 0;

    // ---- w[idx] pooled sum: lane-parallel gather + wave32 xor reduction ----
    float wsum = ((lane < L) ? w[i0] : 0.0f) + ((lane + 32 < L) ? w[i1] : 0.0f);
#pragma unroll
    for (int off = 16; off; off >>= 1) wsum += __shfl_xor(wsum, off, 32);

    // ---- main gather/accumulate: lane owns columns {2*lane, 2*lane+1}.
    // Fully unrolled (L is compile-time): many global_load_b64 in flight per wave,
    // shfl lane index is an immediate -> v_readlane_b32.
    float sx = 0.f, sy = 0.f, qx = 0.f, qy = 0.f;
#pragma unroll
    for (int l = 0; l < L; ++l) {
        const int j = __shfl((l < 32) ? i0 : i1, l & 31, 32);
        const float2 e = reinterpret_cast<const float2*>(emb + (size_t)j * DIM)[lane];
        sx += e.x;       sy += e.y;
        qx += e.x * e.x; qy += e.y * e.y;
    }
    const float bix = 0.5f * (sx * sx - qx);
    const float biy = 0.5f * (sy * sy - qy);

    // ---- emit sum-embedding columns of the output row (stride 66 floats) ----
    float* orow = out + (size_t)row * OUTW;           // 8-byte aligned (264*row)
    reinterpret_cast<float2*>(orow)[lane] = make_float2(sx, sy);

    lds_bi[wv][lane * 2]     = bix;
    lds_bi[wv][lane * 2 + 1] = biy;
    if (lane == 0) lds_ws[wv] = wsum;

    __syncthreads();

    // ---- wave 0: 16 row-dots bi·h1 via V_WMMA_F32_16X16X4_F32 ----
    if (wv == 0) {
        // A 16x4 f32 layout: lanes 0-15 hold M=lane, K={0,1} in VGPR0/1;
        //                    lanes 16-31 hold M=lane-16, K={2,3}.
        const int m  = lane & 15;
        const int kb = (lane < 16) ? 0 : 2;
        v8f c = {};
#pragma unroll
        for (int cc = 0; cc < 16; ++cc) {       // K chunks of 4 covering D=64
            const int k = cc * 4 + kb;
            v2f a, b;
            a[0] = lds_bi[m][k];
            a[1] = lds_bi[m][k + 1];
            b[0] = lds_h1[k];                   // B row K replicated across all N cols
            b[1] = lds_h1[k + 1];
            c = __builtin_amdgcn_wmma_f32_16x16x4_f32(
                    false, a, false, b, (short)0, c, false, false);
        }
        // D layout: lane 0 holds N=0, M=0..7 in c[0..7]; lane 16 holds M=8..15.
        if ((lane & 15) == 0) {
            const int mbase = (lane >> 4) << 3;    // 0 or 8
#pragma unroll
            for (int v = 0; v < 8; ++v) {
                const int mrow = mbase + v;
                const float sc = c[v] + lds_ws[mrow];
                float* trow = out + (size_t)(blockIdx.x * 16 + mrow) * OUTW;
                if (MODE == 0) { trow[DIM] = sc + bias0; trow[DIM + 1] = 1.0f; }
                else           { trow[DIM] = 1.0f;       trow[DIM + 1] = sc;   }
            }
        }
    }
}

__global__ void copy_h2_kernel(const float* __restrict__ h2, float* __restrict__ dst)
{
    dst[threadIdx.x] = h2[threadIdx.x];
}

extern "C" void kernel_launch(void* const* d_in, const int* in_sizes, int n_in,
                              void* d_out, int out_size, void* d_ws, size_t ws_size,
                              hipStream_t stream)
{
    const float* uemb = (const float*)d_in[0];
    const float* iemb = (const float*)d_in[1];
    const float* uw   = (const float*)d_in[2];
    const float* iw   = (const float*)d_in[3];
    const float* gb   = (const float*)d_in[4];
    const float* h1   = (const float*)d_in[5];
    const float* h2   = (const float*)d_in[6];
    const int*   uidx = (const int*)d_in[7];
    const int*   iidx = (const int*)d_in[8];

    constexpr int LU = 50, LI = 20;
    const int NU = in_sizes[7] / LU;     // 4096
    const int NI = in_sizes[8] / LI;     // 16384

    float* P = (float*)d_out;
    float* Q = P + (size_t)NU * OUTW;
    float* H = Q + (size_t)NI * OUTW;

    fm_pool_kernel<LU, 0><<<NU / 16, 512, 0, stream>>>(uemb, uw, uidx, h1, gb, P);
    fm_pool_kernel<LI, 1><<<NI / 16, 512, 0, stream>>>(iemb, iw, iidx, h1, gb, Q);
    copy_h2_kernel<<<1, DIM, 0, stream>>>(h2, H);
}